// GaussianScene_39539468927814
// MI455X (gfx1250) — compile-verified
//
#include <hip/hip_runtime.h>

// ---------------------------------------------------------------------------
// 2D Gaussian splatting for MI455X (gfx1250, wave32).
//
//  q(p,g) is a rank-6 bilinear form -> computed 16 gaussians x 16 pixels at a
//  time with two chained V_WMMA_F32_16X16X4_F32 (K = 8, fp32-exact).
//  Opacity sigmoid(sigmoid(.)) and the tile-cull are folded into the constant
//  feature so alpha = exp(-0.5*q') directly.  Front-to-back compositing with
//  the reference's early-stop is a two-level wave32 scan (8 local steps +
//  one shfl_xor(16) cross-half combine).
// ---------------------------------------------------------------------------

typedef float v2f __attribute__((ext_vector_type(2)));
typedef float v8f __attribute__((ext_vector_type(8)));

#define IMG   256
#define TS    16
#define NTILE 15            // (256-1)/16 tiles per side -> 240x240 rendered
#define NPTS  8192
#define MINW  1e-6f

__global__ __launch_bounds__(256) void splat_zero(float* __restrict__ out, int n) {
    int i = blockIdx.x * blockDim.x + threadIdx.x;
    if (i < n) out[i] = 0.0f;
}

// block = 256 threads = 8 waves; grid = (15, 15, 2).
// wave w of block (bx,by,bz) renders image row  py = by*16 + bz*8 + w,
// pixels px = bx*16 .. bx*16+15.  Waves are fully independent.
__global__ __launch_bounds__(256) void splat_render(
    const float* __restrict__ pts,   // [8192][2]
    const float* __restrict__ cols,  // [8192][3]
    const float* __restrict__ opac,  // [8192]
    const float* __restrict__ icov,  // [8192][2][2]
    const float* __restrict__ mnx, const float* __restrict__ mxx,
    const float* __restrict__ mny, const float* __restrict__ mxy,
    float* __restrict__ out)         // [256][256][3]
{
    __shared__ float4 lcol[8][16];            // per-wave color broadcast slots

    const int lane = threadIdx.x & 31;
    const int wave = threadIdx.x >> 5;
    const int h    = lane >> 4;               // half of wave (0/1)
    const int n    = lane & 15;               // pixel column / gaussian slot
    const float tx = (float)(blockIdx.x * TS);
    const float ty = (float)(blockIdx.y * TS);
    const int  row = blockIdx.z * 8 + wave;
    const float fpx = tx + (float)n;
    const float fpy = ty + (float)row;

    // B operand (pixel features, K x N=16), f(p) = [px^2, px*py, py^2, px, py, 1, 0, 0]
    // assumed layout: lanes 0-15 hold {K0,K1} (N=lane), lanes 16-31 hold {K2,K3}.
    v2f Blo, Bhi;
    if (h == 0) { Blo.x = fpx * fpx; Blo.y = fpx * fpy; Bhi.x = fpy;  Bhi.y = 1.0f; }
    else        { Blo.x = fpy * fpy; Blo.y = fpx;       Bhi.x = 0.0f; Bhi.y = 0.0f; }

    float T = 1.0f, aR = 0.0f, aG = 0.0f, aB = 0.0f;

    for (int base = 0; base < NPTS; base += 16) {
        const int g = base + n;               // gaussian this lane stages

        // tile-vs-bbox cull (exact: culled gaussians have alpha==0 in reference)
        bool inTile = (mnx[g] <= tx + (float)TS) && (mxx[g] >= tx) &&
                      (mny[g] <= ty + (float)TS) && (mxy[g] >= ty);
        if (__ballot(inTile) == 0ull) continue;   // whole batch misses the tile

        __builtin_prefetch(icov + 4 * (g + 16), 0, 1);   // global_prefetch_b8

        const float4 ic = *(const float4*)(icov + 4 * g);
        const float gx = pts[2 * g], gy = pts[2 * g + 1];

        // sigma2 = sigmoid(sigmoid(opac))
        const float s1 = 1.0f / (1.0f + __expf(-opac[g]));
        const float sg = 1.0f / (1.0f + __expf(-s1));

        const float A_ = ic.x, BC = ic.y + ic.z, D_ = ic.w;
        const float L3 = -2.0f * A_ * gx - BC * gy;
        const float L4 = -BC * gx - 2.0f * D_ * gy;
        float C5 = A_ * gx * gx + BC * gx * gy + D_ * gy * gy - 2.0f * __logf(sg);
        if (!inTile) C5 = 1e30f;              // exp(-0.5e30) == 0 -> alpha = 0

        // broadcast colors through LDS (per-wave slot, wave-synchronous)
        if (h == 0)
            lcol[wave][n] = make_float4(cols[3 * g], cols[3 * g + 1], cols[3 * g + 2], 0.0f);
        asm volatile("s_wait_dscnt 0" ::: "memory");

        // A operand (gaussian features, M=16 x K): lanes 0-15 {K0,K1}=(A_,BC),
        // lanes 16-31 {K2,K3}=(D_,L3);  hi: {K4,K5}=(L4,C5) / {K6,K7}=0.
        v2f Alo, Ahi;
        if (h == 0) { Alo.x = A_; Alo.y = BC; Ahi.x = L4;  Ahi.y = C5;  }
        else        { Alo.x = D_; Alo.y = L3; Ahi.x = 0.f; Ahi.y = 0.f; }

        // q' = A x B  (+0), two x4-f32 WMMAs chain K=8
        v8f q = {};
        q = __builtin_amdgcn_wmma_f32_16x16x4_f32(false, Alo, false, Blo,
                                                  (short)0, q, false, false);
        q = __builtin_amdgcn_wmma_f32_16x16x4_f32(false, Ahi, false, Bhi,
                                                  (short)0, q, false, false);
        // lane now holds q' for pixel n, gaussians m = 8*h + r  (r = VGPR index)

        float al[8], lp = 1.0f;
        #pragma unroll
        for (int r = 0; r < 8; ++r) {
            al[r] = __expf(-0.5f * q[r]);     // alpha (opacity+cull pre-folded)
            lp   *= (1.0f - al[r]);           // local half-product of (1-alpha)
        }
        const float pp = __shfl_xor(lp, 16, 32);     // partner half's product
        float t = h ? T * pp : T;             // globally-correct T entering half

        #pragma unroll
        for (int r = 0; r < 8; ++r) {
            const float tn = t * (1.0f - al[r]);
            // reference: contrib_j = 0 once T_{j+1} < MIN (T is monotone)
            const float c  = (tn < MINW) ? 0.0f : t * al[r];
            const float4 cv = lcol[wave][h * 8 + r];
            aR += c * cv.x; aG += c * cv.y; aB += c * cv.z;
            t = tn;
        }
        T *= lp * pp;                         // same on both halves

        if (__ballot(T >= MINW) == 0ull) break;   // every pixel saturated
    }

    // combine the two half-accumulators and store (lanes 0-15 own the pixels)
    aR += __shfl_xor(aR, 16, 32);
    aG += __shfl_xor(aG, 16, 32);
    aB += __shfl_xor(aB, 16, 32);
    if (h == 0) {
        const int px = blockIdx.x * TS + n;
        const int py = blockIdx.y * TS + row;
        float* o = out + ((size_t)px * IMG + py) * 3;
        o[0] = aR; o[1] = aG; o[2] = aB;
    }
}

extern "C" void kernel_launch(void* const* d_in, const int* in_sizes, int n_in,
                              void* d_out, int out_size, void* d_ws, size_t ws_size,
                              hipStream_t stream) {
    (void)in_sizes; (void)n_in; (void)d_ws; (void)ws_size; (void)out_size;
    const float* pts  = (const float*)d_in[0];
    const float* cols = (const float*)d_in[1];
    const float* opa  = (const float*)d_in[2];
    const float* icv  = (const float*)d_in[3];
    const float* mnx  = (const float*)d_in[4];
    const float* mxx  = (const float*)d_in[5];
    const float* mny  = (const float*)d_in[6];
    const float* mxy  = (const float*)d_in[7];
    float* out = (float*)d_out;

    const int total = IMG * IMG * 3;
    splat_zero<<<(total + 255) / 256, 256, 0, stream>>>(out, total);

    dim3 grid(NTILE, NTILE, 2);
    splat_render<<<grid, 256, 0, stream>>>(pts, cols, opa, icv,
                                           mnx, mxx, mny, mxy, out);
}